// ContrastiveLoss_27092653703365
// MI455X (gfx1250) — compile-verified
//
#include <hip/hip_runtime.h>

// ---------------------------------------------------------------------------
// ContrastiveLoss on MI455X (gfx1250).
// Memory-bound: image_features = 138 MB read once (~6 us @ 23.3 TB/s floor).
// Dot products + image-row norms computed with V_WMMA_F32_16X16X4_F32
// (full fp32 precision): diagonal of A[16x4] x B[4x16] accumulates per-row
// dots; wmma(a,a) diagonal accumulates squared row norms for free.
// ---------------------------------------------------------------------------

typedef __attribute__((ext_vector_type(2))) float v2f;
typedef __attribute__((ext_vector_type(8))) float v8f;

#define BS   256
#define FTN  8
#define NDIM 512
#define ATN  (BS + FTN)   // 264

// C/D 16x16 f32 layout: VGPR r holds M=r (lanes 0-15, N=lane) and M=r+8
// (lanes 16-31, N=lane-16).  Diagonal D[j,j]:
//   j <  8 : VGPR j,   lane j
//   j >= 8 : VGPR j-8, lane j+16   (i.e. VGPR r at lane r+24)
__device__ __forceinline__ float wmma_diag(const v8f& c, int lane) {
  float d = 0.0f;
#pragma unroll
  for (int r = 0; r < 8; ++r) {
    float v = c[r];
    if (lane == r || lane == r + 24) d = v;
  }
  return d;
}

// One wave per (image b, 16-wide tile of text indices a0..a0+15).
// Per K=4 chunk: A operand lane L holds img[b, a0+(L&15), k0+2*(L>>4)+{0,1}],
// B operand lane L holds txt[a0+(L&15),  k0+2*(L>>4)+{0,1}]  (same pattern,
// since the B 4x16 layout mirrors the A 16x4 layout).  Accumulate:
//   c_dot += A x B       -> diag = dot(img_row_j, txt_row_j)
//   c_ni  += A x A       -> diag = |img_row_j|^2
//   c_nt  += B x B       -> diag = |txt_row_j|^2
__global__ __launch_bounds__(32)
void cl_rand_logits(const float* __restrict__ img,
                    const float* __restrict__ txt,
                    const float* __restrict__ ls,
                    float* __restrict__ logits) {
  const int b    = blockIdx.x;
  const int a0   = blockIdx.y * 16;
  const int lane = threadIdx.x;          // 0..31, full wave32
  const int m    = lane & 15;
  const int half = lane >> 4;

  const float* arow = img + ((size_t)b * ATN + a0 + m) * NDIM + 2 * half;
  const float* brow = txt + (size_t)(a0 + m) * NDIM + 2 * half;

  v8f c_dot = {}; v8f c_ni = {}; v8f c_nt = {};
#pragma unroll 4
  for (int k0 = 0; k0 < NDIM; k0 += 4) {
    v2f av = *(const v2f*)(arow + k0);
    v2f bv = *(const v2f*)(brow + k0);
    c_dot = __builtin_amdgcn_wmma_f32_16x16x4_f32(false, av, false, bv,
                                                  (short)0, c_dot, false, false);
    c_ni  = __builtin_amdgcn_wmma_f32_16x16x4_f32(false, av, false, av,
                                                  (short)0, c_ni,  false, false);
    c_nt  = __builtin_amdgcn_wmma_f32_16x16x4_f32(false, bv, false, bv,
                                                  (short)0, c_nt,  false, false);
  }

  const float d  = wmma_diag(c_dot, lane);
  const float ni = wmma_diag(c_ni,  lane);
  const float nt = wmma_diag(c_nt,  lane);

  if (lane < 8 || lane >= 24) {
    const int j = (lane < 8) ? lane : lane - 16;   // note: m == j here
    const float scale = expf(ls[0]);
    logits[(size_t)b * ATN + a0 + j] = scale * d / sqrtf(ni * nt);
  }
}

// Same structure for the 2048 false-text dot products: flattened row
// fi = t*16 + m maps to img row (b*ATN + BS + f) with b=fi>>3, f=fi&7.
__global__ __launch_bounds__(32)
void cl_false_logits(const float* __restrict__ img,
                     const float* __restrict__ ftxt,
                     const float* __restrict__ ls,
                     float* __restrict__ logits) {
  const int t    = blockIdx.x;           // 0..127
  const int lane = threadIdx.x;
  const int m    = lane & 15;
  const int half = lane >> 4;
  const int fi   = t * 16 + m;           // 0..2047
  const int b    = fi >> 3;
  const int f    = fi & 7;

  const float* arow = img  + ((size_t)b * ATN + BS + f) * NDIM + 2 * half;
  const float* brow = ftxt + (size_t)fi * NDIM + 2 * half;

  v8f c_dot = {}; v8f c_ni = {}; v8f c_nt = {};
#pragma unroll 4
  for (int k0 = 0; k0 < NDIM; k0 += 4) {
    v2f av = *(const v2f*)(arow + k0);
    v2f bv = *(const v2f*)(brow + k0);
    c_dot = __builtin_amdgcn_wmma_f32_16x16x4_f32(false, av, false, bv,
                                                  (short)0, c_dot, false, false);
    c_ni  = __builtin_amdgcn_wmma_f32_16x16x4_f32(false, av, false, av,
                                                  (short)0, c_ni,  false, false);
    c_nt  = __builtin_amdgcn_wmma_f32_16x16x4_f32(false, bv, false, bv,
                                                  (short)0, c_nt,  false, false);
  }

  const float d  = wmma_diag(c_dot, lane);
  const float ni = wmma_diag(c_ni,  lane);
  const float nt = wmma_diag(c_nt,  lane);

  if (lane < 8 || lane >= 24) {
    // diag lane: m == j, so fi/b/f already index the right row
    const float scale = expf(ls[0]);
    logits[(size_t)b * ATN + BS + f] = scale * d / sqrtf(ni * nt);
  }
}

__device__ __forceinline__ float wave_max(float v) {
#pragma unroll
  for (int off = 16; off > 0; off >>= 1)
    v = fmaxf(v, __shfl_xor(v, off, 32));
  return v;
}
__device__ __forceinline__ float wave_sum(float v) {
#pragma unroll
  for (int off = 16; off > 0; off >>= 1)
    v += __shfl_xor(v, off, 32);
  return v;
}

// Per-image CE: one wave per row of logits[256][264]; label = row index.
__global__ __launch_bounds__(32)
void cl_ce_image(const float* __restrict__ logits, float* __restrict__ ce) {
  const int row  = blockIdx.x;
  const int lane = threadIdx.x;
  const float* r = logits + (size_t)row * ATN;

  float mx = -3.402823466e38f;
  for (int i = lane; i < ATN; i += 32) mx = fmaxf(mx, r[i]);
  mx = wave_max(mx);

  float s = 0.0f;
  for (int i = lane; i < ATN; i += 32) s += expf(r[i] - mx);
  s = wave_sum(s);

  if (lane == 0) ce[row] = (logf(s) + mx) - r[row];
}

// Per-text CE: one wave per column of logits[:, :256]; label = column index.
__global__ __launch_bounds__(32)
void cl_ce_text(const float* __restrict__ logits, float* __restrict__ ce) {
  const int col  = blockIdx.x;
  const int lane = threadIdx.x;

  float mx = -3.402823466e38f;
  for (int j = lane; j < BS; j += 32)
    mx = fmaxf(mx, logits[(size_t)j * ATN + col]);
  mx = wave_max(mx);

  float s = 0.0f;
  for (int j = lane; j < BS; j += 32)
    s += expf(logits[(size_t)j * ATN + col] - mx);
  s = wave_sum(s);

  if (lane == 0) ce[col] = (logf(s) + mx) - logits[(size_t)col * ATN + col];
}

// Deterministic final reduction: loss = (mean(ce_img) + mean(ce_txt)) / 2.
__global__ __launch_bounds__(32)
void cl_finalize(const float* __restrict__ ce_img,
                 const float* __restrict__ ce_txt,
                 float* __restrict__ out) {
  const int lane = threadIdx.x;
  float s = 0.0f;
  for (int i = lane; i < BS; i += 32) s += ce_img[i] + ce_txt[i];
  s = wave_sum(s);
  if (lane == 0) out[0] = s / (2.0f * BS);
}

extern "C" void kernel_launch(void* const* d_in, const int* in_sizes, int n_in,
                              void* d_out, int out_size, void* d_ws, size_t ws_size,
                              hipStream_t stream) {
  const float* img  = (const float*)d_in[0];   // [256,264,512]
  const float* rtxt = (const float*)d_in[1];   // [256,512]
  const float* ftxt = (const float*)d_in[2];   // [2048,512]
  const float* ls   = (const float*)d_in[3];   // [1]

  float* ws     = (float*)d_ws;
  float* logits = ws;                 // 256*264 floats
  float* ce_img = logits + BS * ATN;  // 256 floats
  float* ce_txt = ce_img + BS;        // 256 floats

  cl_rand_logits <<<dim3(BS, 16), 32, 0, stream>>>(img, rtxt, ls, logits);
  cl_false_logits<<<dim3(BS * FTN / 16), 32, 0, stream>>>(img, ftxt, ls, logits);
  cl_ce_image    <<<dim3(BS), 32, 0, stream>>>(logits, ce_img);
  cl_ce_text     <<<dim3(BS), 32, 0, stream>>>(logits, ce_txt);
  cl_finalize    <<<dim3(1), 32, 0, stream>>>(ce_img, ce_txt, (float*)d_out);
}